// LINEMF_56556129354266
// MI455X (gfx1250) — compile-verified
//
#include <hip/hip_runtime.h>

// LINE-MF log-scaled matrix: out = log(max(X * (sum(d)/5) * dinv_i * dinv_j, 1e-3))
// Pure streaming elementwise op: 512 MB traffic -> ~22us floor at 23.3 TB/s.
// Uses CDNA5 async global->LDS copies (ASYNCcnt) for double-buffered row prefetch.
// NOTE: linemf_main is defined FIRST so the device-asm snippet shows it.

#define NN 8192
#define BNEG 5

typedef float v4f __attribute__((ext_vector_type(4)));
typedef int   v4i __attribute__((ext_vector_type(4)));

#if defined(__has_builtin)
#  if __has_builtin(__builtin_amdgcn_global_load_async_to_lds_b128)
#    define USE_ASYNC 1
#  else
#    define USE_ASYNC 0
#  endif
#else
#  define USE_ASYNC 0
#endif

__device__ __forceinline__ float elem(float x, float rsc, float dj) {
  // ln(max(x*rsc*dj, 1e-3)); __logf -> v_log_f32 (log2) * ln2
  return __logf(fmaxf(x * rsc * dj, 0.001f));
}

#if USE_ASYNC
typedef __attribute__((address_space(1))) v4i* gas_v4i_ptr;  // global int4*
typedef __attribute__((address_space(3))) v4i* las_v4i_ptr;  // LDS int4*

// Each wave asynchronously stages its own 4KB segment of a 32KB row into LDS.
// 8 x b128 per lane = 8 ASYNCcnt increments per wave.
__device__ __forceinline__ void issue_row(const float* __restrict__ Xrow,
                                          float* ldsbuf, int wave, int lane) {
#pragma unroll
  for (int k = 0; k < 8; ++k) {
    const int c = (wave * 256 + k * 32 + lane) * 4;  // float index, 16B aligned
    __builtin_amdgcn_global_load_async_to_lds_b128(
        (gas_v4i_ptr)(Xrow + c),
        (las_v4i_ptr)(ldsbuf + c),
        0, 0);
  }
}
#endif

// ---------------- main streaming kernel (emitted first) ----------------
__global__ __launch_bounds__(256) void linemf_main(const float* __restrict__ X,
                                                   float* __restrict__ out,
                                                   const float* __restrict__ dinv,
                                                   const float* __restrict__ scale) {
  const int lane = threadIdx.x & 31;
  const int wave = threadIdx.x >> 5;

  // Column scale factors: this thread touches the same 8 float4 columns in
  // every row -> load them once into registers (served from L2 after 1st WG).
  v4f dj[8];
  const v4f* dinv4 = (const v4f*)dinv;
#pragma unroll
  for (int k = 0; k < 8; ++k) dj[k] = dinv4[wave * 256 + k * 32 + lane];

  const float sc = scale[0];

#if USE_ASYNC
  __shared__ float buf[2][NN];  // 2 x 32KB row buffers
  int r = blockIdx.x;
  const int rs = gridDim.x;
  issue_row(X + (size_t)r * NN, buf[0], wave, lane);
  int bi = 0;
  for (; r < NN; r += rs) {
    const int rn = r + rs;
    if (rn < NN) {
      issue_row(X + (size_t)rn * NN, buf[bi ^ 1], wave, lane);
      // 16 outstanding, oldest 8 (current row) must be complete
      asm volatile("s_wait_asynccnt 0x8" ::: "memory");
    } else {
      asm volatile("s_wait_asynccnt 0x0" ::: "memory");
    }

    const float rsc = sc * dinv[r];  // uniform per row
    const v4f* src = (const v4f*)buf[bi];
    v4f* dst = (v4f*)(out + (size_t)r * NN);
#pragma unroll
    for (int k = 0; k < 8; ++k) {
      const int c = wave * 256 + k * 32 + lane;
      const v4f x = src[c];  // ds_load_b128 from this wave's own segment
      v4f o;
      o.x = elem(x.x, rsc, dj[k].x);
      o.y = elem(x.y, rsc, dj[k].y);
      o.z = elem(x.z, rsc, dj[k].z);
      o.w = elem(x.w, rsc, dj[k].w);
      __builtin_nontemporal_store(o, dst + c);  // single-use output stream
    }
    // WAR: LDS reads must finish before this buffer is re-filled next iter.
    asm volatile("s_wait_dscnt 0x0" ::: "memory");
    bi ^= 1;
  }
#else
  // Fallback: direct nontemporal B128 streaming (no LDS staging).
  for (int r = blockIdx.x; r < NN; r += gridDim.x) {
    const float rsc = sc * dinv[r];
    const v4f* src = (const v4f*)(X + (size_t)r * NN);
    v4f* dst = (v4f*)(out + (size_t)r * NN);
#pragma unroll
    for (int k = 0; k < 8; ++k) {
      const int c = wave * 256 + k * 32 + lane;
      const v4f x = __builtin_nontemporal_load(src + c);
      v4f o;
      o.x = elem(x.x, rsc, dj[k].x);
      o.y = elem(x.y, rsc, dj[k].y);
      o.z = elem(x.z, rsc, dj[k].z);
      o.w = elem(x.w, rsc, dj[k].w);
      __builtin_nontemporal_store(o, dst + c);
    }
  }
#endif
}

// ---------------- prep: dinv[] and scale = sum(d)/BNEG ----------------
__global__ __launch_bounds__(256) void linemf_prep(const float* __restrict__ d_rt,
                                                   float* __restrict__ dinv,
                                                   float* __restrict__ scale) {
  __shared__ float red[256];
  const int t = threadIdx.x;
  float s = 0.0f;
  for (int i = t; i < NN; i += 256) {
    const float d = d_rt[i];
    dinv[i] = 1.0f / d;
    s += d;
  }
  red[t] = s;
  __syncthreads();
  for (int off = 128; off > 0; off >>= 1) {
    if (t < off) red[t] += red[t + off];
    __syncthreads();
  }
  if (t == 0) scale[0] = red[0] * (1.0f / (float)BNEG);
}

extern "C" void kernel_launch(void* const* d_in, const int* in_sizes, int n_in,
                              void* d_out, int out_size, void* d_ws, size_t ws_size,
                              hipStream_t stream) {
  (void)in_sizes; (void)n_in; (void)out_size; (void)ws_size;
  const float* X    = (const float*)d_in[0];
  const float* d_rt = (const float*)d_in[1];
  float* out   = (float*)d_out;
  float* dinv  = (float*)d_ws;        // NN floats
  float* scale = dinv + NN;           // 1 float

  linemf_prep<<<1, 256, 0, stream>>>(d_rt, dinv, scale);
  linemf_main<<<1024, 256, 0, stream>>>(X, out, dinv, scale);
}